// GeneralLPModel_85856396248060
// MI455X (gfx1250) — compile-verified
//
#include <hip/hip_runtime.h>
#include <hip/hip_bf16.h>
#include <math.h>

typedef float v2f __attribute__((ext_vector_type(2)));
typedef float v8f __attribute__((ext_vector_type(8)));

#define NCLS 40    // classes per node (row length)
#define WPAD 48    // padded W column count in LDS (zero cols 40..47)

// ---------------------------------------------------------------------------
// 1) Per-node inverse L2 norm: inv[i] = 1 / (||x_i|| + 1e-15)
// ---------------------------------------------------------------------------
__global__ __launch_bounds__(256)
void invnorm_kernel(const float* __restrict__ x, float* __restrict__ inv, int n) {
    int i = blockIdx.x * blockDim.x + threadIdx.x;
    if (i >= n) return;
    const float4* row = (const float4*)(x + (size_t)i * NCLS);
    float s = 0.f;
#pragma unroll
    for (int j = 0; j < NCLS / 4; ++j) {
        float4 v = row[j];
        s += v.x * v.x + v.y * v.y + v.z * v.z + v.w * v.w;
    }
    inv[i] = 1.0f / (sqrtf(s) + 1e-15f);
}

// ---------------------------------------------------------------------------
// 2) Zero the aggregation buffer (float4 granularity)
// ---------------------------------------------------------------------------
__global__ __launch_bounds__(256)
void zero_kernel(float4* __restrict__ p, int n4) {
    int i = blockIdx.x * blockDim.x + threadIdx.x;
    if (i < n4) p[i] = make_float4(0.f, 0.f, 0.f, 0.f);
}

// ---------------------------------------------------------------------------
// 3) Edge scatter: agg[dst] += x[src] * inv[src]
//    5 threads per edge, 8 channels each (two float4 gathers + 8 f32 atomics).
//    All node tables live in L2 (192 MB); L2 atomic rate is the limiter.
// ---------------------------------------------------------------------------
__global__ __launch_bounds__(256)
void scatter_kernel(const float* __restrict__ x, const float* __restrict__ inv,
                    const int* __restrict__ ei, float* __restrict__ agg,
                    int nEdges) {
    long long gid = (long long)blockIdx.x * blockDim.x + threadIdx.x;
    long long total = (long long)nEdges * (NCLS / 8);
    if (gid >= total) return;
    int e  = (int)(gid / (NCLS / 8));
    int g  = (int)(gid % (NCLS / 8));
    int c0 = g * 8;
    int src = ei[e];
    int dst = ei[nEdges + e];
    float s = inv[src];
    const float4* xs = (const float4*)(x + (size_t)src * NCLS + c0);
    float4 a = xs[0];
    float4 b = xs[1];
    float* ad = agg + (size_t)dst * NCLS + c0;
    atomicAdd(ad + 0, a.x * s);
    atomicAdd(ad + 1, a.y * s);
    atomicAdd(ad + 2, a.z * s);
    atomicAdd(ad + 3, a.w * s);
    atomicAdd(ad + 4, b.x * s);
    atomicAdd(ad + 5, b.y * s);
    atomicAdd(ad + 6, b.z * s);
    atomicAdd(ad + 7, b.w * s);
}

// ---------------------------------------------------------------------------
// 4) out = relu(A @ W) via V_WMMA_F32_16X16X4_F32.
//    One wave owns a 16-row tile and keeps 3 column-tile accumulators
//    (cols 0-15, 16-31, 32-47; 40..47 zero-padded).
//
//    W is staged in LDS *K-pair interleaved*: sW2[p][n] = {W[2p][n], W[2p+1][n]}
//    so each B fragment is ONE aligned ds_load_b64 straight into an even
//    VGPR pair (no repacking movs, no EXEC toggles). K=40 -> 10 steps x 3
//    tiles = 30 v_wmma per wave.
//
//    Fragment layouts (ISA 7.12.2, wave32):
//      A 16x4 f32 : lanes 0-15 -> M, reg0 = K {k0 | k0+2}, reg1 = {k0+1 | k0+3}
//      B 4x16 f32 : lanes 0-15 -> N, regs/halves carry K (mirror of A)
//      D 16x16 f32: VGPR r -> M = r (lanes 0-15) / r+8 (lanes 16-31), N = lane%16
//    Bank check: halves read pair-rows p and p+1 -> byte addresses differ by
//    48*8 = 384 B (mod 256 B = +32 banks): lanes 0-15 cover banks 0-31,
//    lanes 16-31 cover banks 32-63. Conflict-free.
// ---------------------------------------------------------------------------
__global__ __launch_bounds__(256)
void gemm_relu_wmma_kernel(const float* __restrict__ A,
                           const float* __restrict__ W,
                           float* __restrict__ out, int n) {
    __shared__ v2f sW2[(NCLS / 2) * WPAD];   // 20 x 48 float2 = 7.5 KB
    for (int i = threadIdx.x; i < (NCLS / 2) * WPAD; i += 256) {
        int p = i / WPAD, c = i % WPAD;
        v2f w;
        w.x = (c < NCLS) ? W[(2 * p + 0) * NCLS + c] : 0.f;
        w.y = (c < NCLS) ? W[(2 * p + 1) * NCLS + c] : 0.f;
        sW2[i] = w;
    }
    __syncthreads();

    const int wave = threadIdx.x >> 5;
    const int lane = threadIdx.x & 31;
    const int tile = blockIdx.x * 8 + wave;
    const int numTiles = (n + 15) >> 4;
    if (tile >= numTiles) return;           // wave-uniform: EXEC stays all-ones

    const int half = lane >> 4;             // which 16-lane half
    const int l16  = lane & 15;

    int rowA = tile * 16 + l16;
    if (rowA > n - 1) rowA = n - 1;         // clamp (no divergence)
    const float* arow = A + (size_t)rowA * NCLS;
    const v2f*   wb   = sW2 + half * WPAD + l16;  // this half's pair-row base

    v8f acc0 = {};
    v8f acc1 = {};
    v8f acc2 = {};

#pragma unroll
    for (int k0 = 0; k0 < NCLS; k0 += 4) {
        const int k = k0 + 2 * half;        // this half's K pair
        v2f a;
        a.x = arow[k];
        a.y = arow[k + 1];                  // merged global_load_b64

        const v2f* wrow = wb + (k0 / 2) * WPAD;  // pair-row p = k0/2 + half

        v2f b0 = wrow[0];                    // cols 0..15  (ds_load_b64)
        acc0 = __builtin_amdgcn_wmma_f32_16x16x4_f32(
            false, a, false, b0, (short)0, acc0, false, false);

        v2f b1 = wrow[16];                   // cols 16..31
        acc1 = __builtin_amdgcn_wmma_f32_16x16x4_f32(
            false, a, false, b1, (short)0, acc1, false, false);

        v2f b2 = wrow[32];                   // cols 32..47 (40..47 zeros)
        acc2 = __builtin_amdgcn_wmma_f32_16x16x4_f32(
            false, a, false, b2, (short)0, acc2, false, false);
    }

    // Store D with ReLU; mask padded columns and out-of-range rows.
#pragma unroll
    for (int r = 0; r < 8; ++r) {
        const int m   = r + 8 * half;
        const int row = tile * 16 + m;
        if (row >= n) continue;
        float* orow = out + (size_t)row * NCLS;
        orow[l16]      = fmaxf(acc0[r], 0.f);
        orow[16 + l16] = fmaxf(acc1[r], 0.f);
        const int n2 = 32 + l16;
        if (n2 < NCLS) orow[n2] = fmaxf(acc2[r], 0.f);
    }
}

// ---------------------------------------------------------------------------
// 5) Row softmax (40 classes), in place.
// ---------------------------------------------------------------------------
__global__ __launch_bounds__(256)
void softmax_kernel(float* __restrict__ x, int n) {
    int i = blockIdx.x * blockDim.x + threadIdx.x;
    if (i >= n) return;
    float* row = x + (size_t)i * NCLS;
    float v[NCLS];
    float mx = -INFINITY;
#pragma unroll
    for (int j = 0; j < NCLS; ++j) {
        v[j] = row[j];
        mx = fmaxf(mx, v[j]);
    }
    float s = 0.f;
#pragma unroll
    for (int j = 0; j < NCLS; ++j) {
        v[j] = __expf(v[j] - mx);
        s += v[j];
    }
    const float r = 1.f / s;
#pragma unroll
    for (int j = 0; j < NCLS; ++j) row[j] = v[j] * r;
}

// ---------------------------------------------------------------------------
extern "C" void kernel_launch(void* const* d_in, const int* in_sizes, int n_in,
                              void* d_out, int out_size, void* d_ws, size_t ws_size,
                              hipStream_t stream) {
    const float* x0 = (const float*)d_in[0];     // [N, 40]
    const float* Ws = (const float*)d_in[1];     // [ITER, 40, 40]
    const int*   ei = (const int*)d_in[2];       // [2, E]
    float* out = (float*)d_out;

    const int N    = in_sizes[0] / NCLS;
    const int E    = in_sizes[2] / 2;
    const int ITER = in_sizes[1] / (NCLS * NCLS);

    // Workspace layout: inv_norm [N] | agg [N*40] | xbuf [N*40]
    char* ws = (char*)d_ws;
    float* inv = (float*)ws;
    size_t off = ((size_t)N * sizeof(float) + 255) & ~(size_t)255;
    float* agg = (float*)(ws + off);
    off += ((size_t)N * NCLS * sizeof(float) + 255) & ~(size_t)255;
    float* xbuf = (float*)(ws + off);

    const int tpb = 256;
    const int nodeBlocks = (N + tpb - 1) / tpb;
    const int n4 = N * NCLS / 4;
    const int zeroBlocks = (n4 + tpb - 1) / tpb;
    const long long scatterThreads = (long long)E * (NCLS / 8);
    const int scatterBlocks = (int)((scatterThreads + tpb - 1) / tpb);
    const int numTiles = (N + 15) / 16;
    const int gemmBlocks = (numTiles + 7) / 8;

    const float* xin = x0;
    for (int it = 0; it < ITER; ++it) {
        float* xnext = (it == ITER - 1) ? out : xbuf;
        invnorm_kernel<<<nodeBlocks, tpb, 0, stream>>>(xin, inv, N);
        zero_kernel<<<zeroBlocks, tpb, 0, stream>>>((float4*)agg, n4);
        scatter_kernel<<<scatterBlocks, tpb, 0, stream>>>(xin, inv, ei, agg, E);
        gemm_relu_wmma_kernel<<<gemmBlocks, tpb, 0, stream>>>(
            agg, Ws + (size_t)it * NCLS * NCLS, xnext, N);
        xin = xnext;
    }
    softmax_kernel<<<nodeBlocks, tpb, 0, stream>>>(out, N);
}